// GAT_89601607729382
// MI455X (gfx1250) — compile-verified
//
#include <hip/hip_runtime.h>
#include <hip/hip_bf16.h>

typedef __attribute__((ext_vector_type(2))) float v2f;
typedef __attribute__((ext_vector_type(8))) float v8f;

#define N_NODES 50000
#define IN_C    128
#define HEADS   8
#define HID     32
#define CAT     256   // HEADS*HID
#define OUT_C   64
#define E_EDGES 800000

// ---------- helpers: order-preserving float<->uint key for atomicMax ----------
__device__ __forceinline__ unsigned fkey(float f) {
    unsigned u = __float_as_uint(f);
    return (u & 0x80000000u) ? ~u : (u | 0x80000000u);
}
__device__ __forceinline__ float keyf(unsigned k) {
    unsigned u = (k & 0x80000000u) ? (k & 0x7FFFFFFFu) : ~k;
    return __uint_as_float(u);
}
__device__ __forceinline__ float lrelu(float v) { return v > 0.f ? v : 0.2f * v; }

// ---------- generic zero ----------
__global__ void zero_f32(float* __restrict__ p, long n) {
    long i = (long)blockIdx.x * blockDim.x + threadIdx.x;
    long stride = (long)gridDim.x * blockDim.x;
    for (; i < n; i += stride) p[i] = 0.f;
}

// ---------- GEMM1: h[N,256] = x[N,128] @ W_gat[128,256], WMMA f32 16x16x4 ----------
// One wave per 16x16 output tile. A: lane(half,r) holds A[r][2*half+{0,1}];
// B: lane(half,r) holds B[2*half+{0,1}][r]; D: vgpr i -> row i+8*half, col r.
__global__ void gemm_xWgat(const float* __restrict__ x, const float* __restrict__ W,
                           float* __restrict__ h) {
    const int wave = blockIdx.x * (blockDim.x >> 5) + (threadIdx.x >> 5);
    const int lane = threadIdx.x & 31;
    const int NT = CAT / 16;                    // 16 column tiles
    const int total = (N_NODES / 16) * NT;      // 50000 tiles
    if (wave >= total) return;
    const int mt = wave / NT, nt = wave % NT;
    const int row0 = mt * 16, col0 = nt * 16;
    const int half = lane >> 4, r = lane & 15;

    v8f acc = {};
    const float* xr = x + (size_t)(row0 + r) * IN_C;
    const float* wc = W + col0 + r;
    for (int k = 0; k < IN_C; k += 4) {
        v2f a, b;
        a.x = xr[k + 2 * half];
        a.y = xr[k + 2 * half + 1];
        b.x = wc[(size_t)(k + 2 * half) * CAT];
        b.y = wc[(size_t)(k + 2 * half + 1) * CAT];
        acc = __builtin_amdgcn_wmma_f32_16x16x4_f32(false, a, false, b,
                                                    (short)0, acc, false, false);
    }
    float* hp = h + (size_t)col0 + r;
#pragma unroll
    for (int i = 0; i < 8; ++i)
        hp[(size_t)(row0 + i + 8 * half) * CAT] = acc[i];
}

// ---------- attention coefficients: a_s/a_d [N,8] ----------
__global__ void attn_dots(const float* __restrict__ h, const float* __restrict__ att_src,
                          const float* __restrict__ att_dst,
                          float* __restrict__ a_s, float* __restrict__ a_d) {
    int gid = blockIdx.x * blockDim.x + threadIdx.x;
    if (gid >= N_NODES * HEADS) return;
    int node = gid >> 3, head = gid & 7;
    const float* hp = h + (size_t)node * CAT + head * HID;
    const float* as = att_src + head * HID;
    const float* ad = att_dst + head * HID;
    float s = 0.f, d = 0.f;
#pragma unroll
    for (int c = 0; c < HID; c += 4) {
        float4 hv = *(const float4*)(hp + c);
        float4 sv = *(const float4*)(as + c);
        float4 dv = *(const float4*)(ad + c);
        s += hv.x * sv.x + hv.y * sv.y + hv.z * sv.z + hv.w * sv.w;
        d += hv.x * dv.x + hv.y * dv.y + hv.z * dv.z + hv.w * dv.w;
    }
    a_s[gid] = s;
    a_d[gid] = d;
}

// ---------- edge pass 1: segment max ----------
__global__ void edge_max(const int* __restrict__ src, const int* __restrict__ dst,
                         const float* __restrict__ a_s, const float* __restrict__ a_d,
                         unsigned* __restrict__ mkey) {
    int e = blockIdx.x * blockDim.x + threadIdx.x;
    if (e >= E_EDGES) return;
    int s = src[e], d = dst[e];
#pragma unroll
    for (int hh = 0; hh < HEADS; ++hh) {
        float v = lrelu(a_s[s * HEADS + hh] + a_d[d * HEADS + hh]);
        atomicMax(&mkey[d * HEADS + hh], fkey(v));
    }
}

// ---------- convert keys back to floats; empty segments -> 0 ----------
__global__ void fix_m(unsigned* __restrict__ mkey, float* __restrict__ m) {
    int i = blockIdx.x * blockDim.x + threadIdx.x;
    if (i >= N_NODES * HEADS) return;
    unsigned k = mkey[i];
    m[i] = (k == 0u) ? 0.f : keyf(k);
}

// ---------- edge pass 2: softmax denominators + degree counts ----------
__global__ void edge_sum(const int* __restrict__ src, const int* __restrict__ dst,
                         const float* __restrict__ a_s, const float* __restrict__ a_d,
                         const float* __restrict__ m, float* __restrict__ denom,
                         float* __restrict__ cnt) {
    int e = blockIdx.x * blockDim.x + threadIdx.x;
    if (e >= E_EDGES) return;
    int s = src[e], d = dst[e];
#pragma unroll
    for (int hh = 0; hh < HEADS; ++hh) {
        float v = lrelu(a_s[s * HEADS + hh] + a_d[d * HEADS + hh]);
        atomicAdd(&denom[d * HEADS + hh], expf(v - m[d * HEADS + hh]));
    }
    atomicAdd(&cnt[d], 1.0f);
}

// ---------- edge pass 3: weighted message scatter into out1 (x32 accumulator) ----------
__global__ void edge_scatter(const int* __restrict__ src, const int* __restrict__ dst,
                             const float* __restrict__ a_s, const float* __restrict__ a_d,
                             const float* __restrict__ m, const float* __restrict__ denom,
                             const float* __restrict__ h, float* __restrict__ out1) {
    int gid = blockIdx.x * blockDim.x + threadIdx.x;
    if (gid >= E_EDGES * HEADS) return;
    int e = gid >> 3, hh = gid & 7;
    int s = src[e], d = dst[e];
    float v = lrelu(a_s[s * HEADS + hh] + a_d[d * HEADS + hh]);
    float alpha = expf(v - m[d * HEADS + hh]) / (denom[d * HEADS + hh] + 1e-16f);
    const float4* hp = (const float4*)(h + (size_t)s * CAT + hh * HID);
    float* op = out1 + (size_t)d * CAT + hh * HID;
#pragma unroll
    for (int j = 0; j < HID / 4; ++j) {
        float4 hv = hp[j];
        atomicAdd(op + 4 * j + 0, hv.x * alpha);
        atomicAdd(op + 4 * j + 1, hv.y * alpha);
        atomicAdd(op + 4 * j + 2, hv.z * alpha);
        atomicAdd(op + 4 * j + 3, hv.w * alpha);
    }
}

// ---------- bias + ReLU in place: out1 -> x32 ----------
__global__ void bias_relu(float* __restrict__ x32, const float* __restrict__ b_gat) {
    int i = blockIdx.x * blockDim.x + threadIdx.x;
    if (i >= N_NODES * CAT) return;
    float v = x32[i] + b_gat[i & (CAT - 1)];
    x32[i] = v > 0.f ? v : 0.f;
}

// ---------- SAGE neighbor sum scatter ----------
__global__ void sage_scatter(const int* __restrict__ src, const int* __restrict__ dst,
                             const float* __restrict__ x32, float* __restrict__ agg) {
    int gid = blockIdx.x * blockDim.x + threadIdx.x;
    if (gid >= E_EDGES * HEADS) return;
    int e = gid >> 3, part = gid & 7;
    int s = src[e], d = dst[e];
    const float4* xp = (const float4*)(x32 + (size_t)s * CAT + part * HID);
    float* op = agg + (size_t)d * CAT + part * HID;
#pragma unroll
    for (int j = 0; j < HID / 4; ++j) {
        float4 xv = xp[j];
        atomicAdd(op + 4 * j + 0, xv.x);
        atomicAdd(op + 4 * j + 1, xv.y);
        atomicAdd(op + 4 * j + 2, xv.z);
        atomicAdd(op + 4 * j + 3, xv.w);
    }
}

// ---------- mean normalize ----------
__global__ void agg_norm(float* __restrict__ agg, const float* __restrict__ cnt) {
    int i = blockIdx.x * blockDim.x + threadIdx.x;
    if (i >= N_NODES * CAT) return;
    int node = i >> 8;
    agg[i] /= fmaxf(cnt[node], 1.0f);
}

// ---------- GEMM2: out2 = agg@W_l + x32@W_r + b_l, fused dual WMMA chain ----------
__global__ void gemm_out2(const float* __restrict__ agg, const float* __restrict__ x32,
                          const float* __restrict__ Wl, const float* __restrict__ Wr,
                          const float* __restrict__ bl, float* __restrict__ out2) {
    const int wave = blockIdx.x * (blockDim.x >> 5) + (threadIdx.x >> 5);
    const int lane = threadIdx.x & 31;
    const int NT = OUT_C / 16;                  // 4 column tiles
    const int total = (N_NODES / 16) * NT;      // 12500 tiles
    if (wave >= total) return;
    const int mt = wave / NT, nt = wave % NT;
    const int row0 = mt * 16, col0 = nt * 16;
    const int half = lane >> 4, r = lane & 15;

    v8f acc = {};
    const float* ar  = agg + (size_t)(row0 + r) * CAT;
    const float* xr  = x32 + (size_t)(row0 + r) * CAT;
    const float* wlc = Wl + col0 + r;
    const float* wrc = Wr + col0 + r;
    for (int k = 0; k < CAT; k += 4) {
        v2f a, b;
        a.x = ar[k + 2 * half];
        a.y = ar[k + 2 * half + 1];
        b.x = wlc[(size_t)(k + 2 * half) * OUT_C];
        b.y = wlc[(size_t)(k + 2 * half + 1) * OUT_C];
        acc = __builtin_amdgcn_wmma_f32_16x16x4_f32(false, a, false, b,
                                                    (short)0, acc, false, false);
        a.x = xr[k + 2 * half];
        a.y = xr[k + 2 * half + 1];
        b.x = wrc[(size_t)(k + 2 * half) * OUT_C];
        b.y = wrc[(size_t)(k + 2 * half + 1) * OUT_C];
        acc = __builtin_amdgcn_wmma_f32_16x16x4_f32(false, a, false, b,
                                                    (short)0, acc, false, false);
    }
    float bias = bl[col0 + r];
#pragma unroll
    for (int i = 0; i < 8; ++i)
        out2[(size_t)(row0 + i + 8 * half) * OUT_C + col0 + r] = acc[i] + bias;
}

extern "C" void kernel_launch(void* const* d_in, const int* in_sizes, int n_in,
                              void* d_out, int out_size, void* d_ws, size_t ws_size,
                              hipStream_t stream) {
    (void)in_sizes; (void)n_in; (void)out_size; (void)ws_size;
    const float* x       = (const float*)d_in[0];
    const int*   eidx    = (const int*)d_in[1];      // [2,E] row-major
    const float* W_gat   = (const float*)d_in[2];
    const float* att_src = (const float*)d_in[3];
    const float* att_dst = (const float*)d_in[4];
    const float* b_gat   = (const float*)d_in[5];
    const float* W_l     = (const float*)d_in[6];
    const float* b_l     = (const float*)d_in[7];
    const float* W_r     = (const float*)d_in[8];

    const int* src = eidx;
    const int* dst = eidx + E_EDGES;

    // d_out: [x32: N*CAT][out2: N*OUT]
    float* x32  = (float*)d_out;
    float* out2 = x32 + (size_t)N_NODES * CAT;

    // workspace layout (floats)
    float*    h     = (float*)d_ws;                          // N*CAT
    float*    a_s   = h + (size_t)N_NODES * CAT;             // N*8
    float*    a_d   = a_s + (size_t)N_NODES * HEADS;         // N*8
    unsigned* mkey  = (unsigned*)(a_d + (size_t)N_NODES * HEADS); // N*8 (reused as float m)
    float*    m     = (float*)mkey;
    float*    denom = m + (size_t)N_NODES * HEADS;           // N*8
    float*    cnt   = denom + (size_t)N_NODES * HEADS;       // N
    float*    agg   = cnt + N_NODES;                         // N*CAT

    const int B = 256;

    // zero: [mkey .. end of agg] is contiguous, plus x32 accumulator region
    long zero_n = (long)N_NODES * HEADS * 2 + N_NODES + (long)N_NODES * CAT;
    zero_f32<<<2048, B, 0, stream>>>(m, zero_n);
    zero_f32<<<2048, B, 0, stream>>>(x32, (long)N_NODES * CAT);

    // GAT linear: 50000 tiles, 8 waves/block
    gemm_xWgat<<<(N_NODES / 16) * (CAT / 16) / 8, B, 0, stream>>>(x, W_gat, h);

    attn_dots<<<(N_NODES * HEADS + B - 1) / B, B, 0, stream>>>(h, att_src, att_dst, a_s, a_d);

    edge_max<<<(E_EDGES + B - 1) / B, B, 0, stream>>>(src, dst, a_s, a_d, mkey);
    fix_m<<<(N_NODES * HEADS + B - 1) / B, B, 0, stream>>>(mkey, m);
    edge_sum<<<(E_EDGES + B - 1) / B, B, 0, stream>>>(src, dst, a_s, a_d, m, denom, cnt);
    edge_scatter<<<(E_EDGES * HEADS + B - 1) / B, B, 0, stream>>>(src, dst, a_s, a_d, m,
                                                                 denom, h, x32);
    bias_relu<<<(N_NODES * CAT + B - 1) / B, B, 0, stream>>>(x32, b_gat);

    sage_scatter<<<(E_EDGES * HEADS + B - 1) / B, B, 0, stream>>>(src, dst, x32, agg);
    agg_norm<<<(N_NODES * CAT + B - 1) / B, B, 0, stream>>>(agg, cnt);

    // fused final GEMMs: 12500 tiles, 8 waves/block -> 1563 blocks (guarded)
    gemm_out2<<<((N_NODES / 16) * (OUT_C / 16) + 7) / 8, B, 0, stream>>>(agg, x32, W_l, W_r,
                                                                         b_l, out2);
}